// QKVAttention_4621384810779
// MI455X (gfx1250) — compile-verified
//
#include <hip/hip_runtime.h>

typedef float v2f __attribute__((ext_vector_type(2)));
typedef float v8f __attribute__((ext_vector_type(8)));

constexpr int kLen    = 2048;          // t / s length
constexpr int kTiles  = kLen / 16;     // 128 16-wide tiles
constexpr int kHeads  = 32;            // bs(4) * n_heads(8)
constexpr int kQKVStride = 192 * kLen; // rows per head-batch in qkv_

__device__ __forceinline__ v8f wmma_f32(v2f a, v2f b, v8f c) {
  // V_WMMA_F32_16X16X4_F32 : D = A(16x4) * B(4x16) + C(16x16), all fp32
  return __builtin_amdgcn_wmma_f32_16x16x4_f32(false, a, false, b,
                                               (short)0, c, false, false);
}

// ---------------------------------------------------------------------------
// Pass 1: per (head, s) column stats of w[t,s] over all t (softmax axis = t)
//   m[s] = max_t w[t,s],  l[s] = sum_t exp(w[t,s] - m[s]),  w = (q.k)/8
// One wave32 per (head, 16-wide s-block); online update over 128 t-tiles.
// ---------------------------------------------------------------------------
__global__ __launch_bounds__(32) void attn_stage1(const float* __restrict__ qkv,
                                                  float* __restrict__ mArr,
                                                  float* __restrict__ lArr) {
  const int head = blockIdx.y;
  const int s0   = blockIdx.x * 16;
  const int lane = threadIdx.x;
  const int half = lane >> 4;   // 0: lanes 0-15, 1: lanes 16-31
  const int col  = lane & 15;
  const float* q = qkv + head * kQKVStride;           // q[c][t]
  const float* k = q + 64 * kLen;                     // k[c][s]

  // B operand (k tile for this s-block), loop-invariant: 16 K-steps of 4.
  v2f kb[16];
#pragma unroll
  for (int kk = 0; kk < 16; ++kk) {
    const int c = 4 * kk + 2 * half;
    kb[kk].x = k[c * kLen + s0 + col];
    kb[kk].y = k[(c + 1) * kLen + s0 + col];
  }

  float m = -__builtin_inff();
  float l = 0.0f;

  for (int tb = 0; tb < kTiles; ++tb) {
    const int t0 = tb * 16;
    v8f acc = {};
#pragma unroll
    for (int kk = 0; kk < 16; ++kk) {
      const int c = 4 * kk + 2 * half;
      v2f a;                                   // A = q^T tile (M=t, K=c)
      a.x = q[c * kLen + t0 + col];
      a.y = q[(c + 1) * kLen + t0 + col];
      acc = wmma_f32(a, kb[kk], acc);
    }
    // D layout: vgpr r holds (t = t0 + r + 8*half, s = s0 + col).
    float w[8];
    float tmax = -__builtin_inff();
#pragma unroll
    for (int r = 0; r < 8; ++r) {
      w[r] = acc[r] * 0.125f;                  // scale^2 = 1/sqrt(ch) = 1/8
      tmax = fmaxf(tmax, w[r]);
    }
    tmax = fmaxf(tmax, __shfl_xor(tmax, 16, 32));   // combine lane L with L+16
    const float mnew = fmaxf(m, tmax);
    float ssum = 0.0f;
#pragma unroll
    for (int r = 0; r < 8; ++r) ssum += __expf(w[r] - mnew);
    ssum += __shfl_xor(ssum, 16, 32);
    l = l * __expf(m - mnew) + ssum;
    m = mnew;
  }
  if (lane < 16) {
    mArr[head * kLen + s0 + lane] = m;
    lArr[head * kLen + s0 + lane] = l;
  }
}

// ---------------------------------------------------------------------------
// Pass 2: a[c,t] = sum_s exp(w[t,s]-m[s])/l[s] * v[c,s]
// One wave32 per (head, 16-wide t-block); recompute w tile, normalize with
// the pass-1 stats, then P(16x16) x V^T(16x64) via 4x4 fp32 WMMAs.
// ---------------------------------------------------------------------------
__global__ __launch_bounds__(32) void attn_stage2(const float* __restrict__ qkv,
                                                  const float* __restrict__ mArr,
                                                  const float* __restrict__ lArr,
                                                  float* __restrict__ out) {
  const int head = blockIdx.y;
  const int t0   = blockIdx.x * 16;
  const int lane = threadIdx.x;
  const int half = lane >> 4;
  const int col  = lane & 15;
  const float* q = qkv + head * kQKVStride;
  const float* k = q + 64 * kLen;
  const float* v = q + 128 * kLen;

  __shared__ float ldsP[16 * 17];   // P tile, D-layout -> A-layout transpose
  __shared__ float ldsV[64 * 17];   // V tile (64c x 16s) staged coalesced

  // A operand (q tile for this t-block), loop-invariant.
  v2f qa[16];
#pragma unroll
  for (int kk = 0; kk < 16; ++kk) {
    const int c = 4 * kk + 2 * half;
    qa[kk].x = q[c * kLen + t0 + col];
    qa[kk].y = q[(c + 1) * kLen + t0 + col];
  }

  v8f oacc[4] = {{}, {}, {}, {}};   // D tiles: M=t-local, N = c in 4 blocks of 16

  for (int sb = 0; sb < kTiles; ++sb) {
    const int s0 = sb * 16;

    // Stage v[0:64][s0:s0+16] into LDS (coalesced 64B row segments).
#pragma unroll
    for (int j = 0; j < 32; ++j) {
      const int idx = j * 32 + lane;
      const int c = idx >> 4, sl = idx & 15;
      ldsV[c * 17 + sl] = v[c * kLen + s0 + sl];
    }

    // Recompute w tile = q^T k / 8.
    v8f wacc = {};
#pragma unroll
    for (int kk = 0; kk < 16; ++kk) {
      const int c = 4 * kk + 2 * half;
      v2f kb;
      kb.x = k[c * kLen + s0 + col];
      kb.y = k[(c + 1) * kLen + s0 + col];
      wacc = wmma_f32(qa[kk], kb, wacc);
    }

    const float ms = mArr[head * kLen + s0 + col];
    const float rl = 1.0f / lArr[head * kLen + s0 + col];
#pragma unroll
    for (int r = 0; r < 8; ++r)
      ldsP[(r + 8 * half) * 17 + col] = __expf(wacc[r] * 0.125f - ms) * rl;
    __syncthreads();

    // oacc[M=t, N=c] += P(M=t, K=s-local) * V^T(K=s-local, N=c)
#pragma unroll
    for (int kk2 = 0; kk2 < 4; ++kk2) {
      const int srow = 4 * kk2 + 2 * half;
      v2f a2;                                  // A-layout read of P
      a2.x = ldsP[col * 17 + srow];
      a2.y = ldsP[col * 17 + srow + 1];
#pragma unroll
      for (int nt = 0; nt < 4; ++nt) {
        v2f vb;                                // B-layout read of V^T
        vb.x = ldsV[(nt * 16 + col) * 17 + srow];
        vb.y = ldsV[(nt * 16 + col) * 17 + srow + 1];
        oacc[nt] = wmma_f32(a2, vb, oacc[nt]);
      }
    }
    __syncthreads();
  }

  // Transpose result through LDS for coalesced stores: out[(head*64+c)*L + t].
#pragma unroll
  for (int nt = 0; nt < 4; ++nt)
#pragma unroll
    for (int r = 0; r < 8; ++r)
      ldsV[(nt * 16 + col) * 17 + r + 8 * half] = oacc[nt][r];
  __syncthreads();
#pragma unroll
  for (int j = 0; j < 32; ++j) {
    const int idx = j * 32 + lane;
    const int c = idx >> 4, tl = idx & 15;
    out[(head * 64 + c) * kLen + t0 + tl] = ldsV[c * 17 + tl];
  }
}

extern "C" void kernel_launch(void* const* d_in, const int* in_sizes, int n_in,
                              void* d_out, int out_size, void* d_ws, size_t ws_size,
                              hipStream_t stream) {
  const float* qkv = (const float*)d_in[0];
  float* out  = (float*)d_out;
  float* mArr = (float*)d_ws;                 // 32*2048 floats
  float* lArr = mArr + kHeads * kLen;         // 32*2048 floats (512 KB total)

  dim3 grid(kTiles, kHeads);                  // 128 x 32 workgroups, 1 wave each
  attn_stage1<<<grid, 32, 0, stream>>>(qkv, mArr, lArr);
  attn_stage2<<<grid, 32, 0, stream>>>(qkv, mArr, lArr, out);
}